// sample_func_89309549953534
// MI455X (gfx1250) — compile-verified
//
#include <hip/hip_runtime.h>
#include <hip/hip_bf16.h>
#include <math.h>

// Problem constants (match reference):
//   B=8 images, R=300 rois, C=21 classes (class 0 = background), K = C = 21
//   NMS threshold 0.3, BBOX_NORMALIZE_STDS = {0.1,0.1,0.2,0.2}
#define NB   8
#define NR   300
#define NC   21
#define NCLS 20            // foreground classes
#define NK   21            // padded detections per image
#define NW   12            // 32-bit mask words per row (padded, 48B = 3x b128)
#define NWU  10            // words actually used (ceil(300/32))
#define NFLAT (NCLS * NR)  // 6000 flattened (class, rank) slots per image

// ---------------------------------------------------------------------------
// Kernel 1: per-(image,class) box decode + stable score ranking + bitmask NMS.
// One block per (b, class-1); 320 threads = 10 wave32s on one WGP.
// Phases: (A) decode+rank+scatter, (B) parallel IoU suppression bitmask,
// (C) single-lane greedy scan with the removal mask in registers.
// Only 4 block barriers total (vs 300 in a naive greedy loop) -- the serial
// NMS dependency chain collapses to a register ALU chain fed by ds_load_b128.
// ---------------------------------------------------------------------------
__global__ __launch_bounds__(320) void decode_rank_nms_kernel(
    const float* __restrict__ cls_prob,   // [B, R, NC]
    const float* __restrict__ rois,       // [B, R, 5]   (col0 unused)
    const float* __restrict__ bbox_pred,  // [B, R, 4*NC]
    const float* __restrict__ im_info,    // [B, 3] = {H, W, scale}
    const float* __restrict__ thr,        // [NC]
    float* __restrict__ ws_box,           // [B, NCLS, R, 4]  sorted boxes
    int*   __restrict__ ws_keep)          // [B, NCLS, R]     keep flags
{
    __shared__ float ss[NR];                          // scores (-inf if invalid)
    __shared__ float sb[NR][4];                       // boxes, score-sorted
    __shared__ float sarea[NR];                       // areas of sorted boxes
    __shared__ int   skeep[NR];                       // valid -> final keep
    __shared__ __align__(16) unsigned smask[NR][NW];  // suppression bit rows

    const int t    = threadIdx.x;
    const int blk  = blockIdx.x;
    const int b    = blk / NCLS;
    const int crel = blk % NCLS;
    const int c    = crel + 1;       // foreground class index 1..20

    float bx0 = 0.f, bx1 = 0.f, bx2 = 0.f, bx3 = 0.f;
    int   valid = 0;

    // ---- Phase A: decode + threshold ------------------------------------
    if (t < NR) {
        const float* roi = rois + ((size_t)b * NR + t) * 5;
        const float x1 = roi[1], y1 = roi[2], x2 = roi[3], y2 = roi[4];
        const float w  = x2 - x1 + 1.0f;
        const float h  = y2 - y1 + 1.0f;
        const float cx = x1 + 0.5f * w;
        const float cy = y1 + 0.5f * h;

        const float* dp = bbox_pred + ((size_t)b * NR + t) * (4 * NC) + 4 * c;
        const float d0 = dp[0] * 0.1f;
        const float d1 = dp[1] * 0.1f;
        const float d2 = dp[2] * 0.2f;
        const float d3 = dp[3] * 0.2f;

        const float pcx = d0 * w + cx;
        const float pcy = d1 * h + cy;
        const float pw  = expf(d2) * w;
        const float ph  = expf(d3) * h;

        const float imh = im_info[b * 3 + 0];
        const float imw = im_info[b * 3 + 1];
        const float xmax = imw - 1.0f, ymax = imh - 1.0f;

        bx0 = fminf(fmaxf(pcx - 0.5f * pw, 0.0f), xmax);
        bx1 = fminf(fmaxf(pcy - 0.5f * ph, 0.0f), ymax);
        bx2 = fminf(fmaxf(pcx + 0.5f * pw, 0.0f), xmax);
        bx3 = fminf(fmaxf(pcy + 0.5f * ph, 0.0f), ymax);

        const float score = cls_prob[((size_t)b * NR + t) * NC + c];
        valid = (score > thr[c]) ? 1 : 0;
        ss[t] = valid ? score : -__builtin_inff();
    }
    __syncthreads();

    // Stable descending rank == jnp.argsort(-sm) (stable sort): the tie-break
    // on index also orders the -inf (invalid) tail exactly like the reference.
    if (t < NR) {
        const float sr = ss[t];
        int rank = 0;
        #pragma unroll 4
        for (int j = 0; j < NR; ++j) {
            const float sj = ss[j];
            rank += ((sj > sr) || (sj == sr && j < t)) ? 1 : 0;
        }
        sb[rank][0] = bx0; sb[rank][1] = bx1; sb[rank][2] = bx2; sb[rank][3] = bx3;
        sarea[rank] = (bx2 - bx0 + 1.0f) * (bx3 - bx1 + 1.0f);
        skeep[rank] = valid;
    }
    __syncthreads();

    // ---- Phase B: pairwise suppression bitmask (fully parallel) ---------
    // Row t, bit j set iff (j > t) && IoU(sorted_t, sorted_j) > 0.3.
    if (t < NR) {
        const float ax1 = sb[t][0], ay1 = sb[t][1];
        const float ax2 = sb[t][2], ay2 = sb[t][3];
        const float aa  = sarea[t];
        for (int w = 0; w < NW; ++w) {
            unsigned bits = 0u;
            const int j0 = w << 5;
            #pragma unroll
            for (int k = 0; k < 32; ++k) {
                const int j = j0 + k;
                if (j < NR) {
                    const float ix1 = fmaxf(ax1, sb[j][0]);
                    const float iy1 = fmaxf(ay1, sb[j][1]);
                    const float ix2 = fminf(ax2, sb[j][2]);
                    const float iy2 = fminf(ay2, sb[j][3]);
                    const float inter = fmaxf(ix2 - ix1 + 1.0f, 0.0f) *
                                        fmaxf(iy2 - iy1 + 1.0f, 0.0f);
                    const float iou = inter / (aa + sarea[j] - inter);
                    if (j > t && iou > 0.3f) bits |= (1u << k);
                }
            }
            smask[t][w] = bits;
        }
    }
    __syncthreads();

    // ---- Phase C: serial greedy scan, one lane, removal mask in VGPRs ---
    if (t == 0) {
        unsigned rem[NWU];
        #pragma unroll
        for (int w = 0; w < NWU; ++w) rem[w] = 0u;
        for (int i = 0; i < NR; ++i) {
            const int w   = i >> 5;
            const int bit = i & 31;
            const int keep = skeep[i] && !((rem[w] >> bit) & 1u);
            skeep[i] = keep;
            const unsigned km = keep ? 0xffffffffu : 0u;  // branchless update
            #pragma unroll
            for (int w2 = 0; w2 < NWU; ++w2) rem[w2] |= smask[i][w2] & km;
        }
    }
    __syncthreads();

    // ---- Store sorted boxes (b128) + final keep flags -------------------
    if (t < NR) {
        const size_t base = (size_t)blk * NR + t;
        ws_keep[base] = skeep[t];
        float4 v;
        v.x = sb[t][0]; v.y = sb[t][1]; v.z = sb[t][2]; v.w = sb[t][3];
        ((float4*)ws_box)[base] = v;
    }
}

// ---------------------------------------------------------------------------
// Kernel 2: per-image selection of the first K kept detections in
// (class asc, score-rank asc) order; zero-pad the rest; emit count.
// Block-wide inclusive scan over the 6000 keep flags (contiguous chunks keep
// the flattened order), staging the 21 output rows in LDS.
// ---------------------------------------------------------------------------
__global__ __launch_bounds__(256) void select_topk_kernel(
    const float* __restrict__ ws_box,   // [B, NFLAT, 4]
    const int*   __restrict__ ws_keep,  // [B, NFLAT]
    float* __restrict__ out,            // [B, NK, 5]
    float* __restrict__ nout)           // [B] (count, stored numerically)
{
    __shared__ int   cnt[256];
    __shared__ float rows[NK * 5];

    const int b = blockIdx.x;
    const int t = threadIdx.x;

    const int CH    = (NFLAT + 255) / 256;   // 24 contiguous slots per thread
    const int start = t * CH;
    const int end   = (start + CH < NFLAT) ? (start + CH) : NFLAT;

    const int*    kp = ws_keep + (size_t)b * NFLAT;
    const float4* bp = (const float4*)(ws_box + (size_t)b * NFLAT * 4);

    int local = 0;
    for (int f = start; f < end; ++f) local += kp[f] ? 1 : 0;
    cnt[t] = local;
    if (t < NK * 5) rows[t] = 0.0f;          // zero padding rows
    __syncthreads();

    // Hillis-Steele inclusive scan over 256 per-thread counts.
    for (int off = 1; off < 256; off <<= 1) {
        const int v   = cnt[t];
        const int add = (t >= off) ? cnt[t - off] : 0;
        __syncthreads();
        cnt[t] = v + add;
        __syncthreads();
    }
    const int excl  = cnt[t] - local;
    const int total = cnt[255];

    int pos = excl;
    if (pos < NK) {
        for (int f = start; f < end && pos < NK; ++f) {
            if (kp[f]) {
                const float4 v = bp[f];           // global_load_b128
                rows[pos * 5 + 0] = v.x;
                rows[pos * 5 + 1] = v.y;
                rows[pos * 5 + 2] = v.z;
                rows[pos * 5 + 3] = v.w;
                rows[pos * 5 + 4] = (float)(f / NR + 1);   // class id
                ++pos;
            }
        }
    }
    __syncthreads();

    if (t < NK * 5) out[(size_t)b * (NK * 5) + t] = rows[t];
    if (t == 0)     nout[b] = (float)(total < NK ? total : NK);
}

// ---------------------------------------------------------------------------
// Host launcher (graph-capture safe: only kernel launches on `stream`).
// ---------------------------------------------------------------------------
extern "C" void kernel_launch(void* const* d_in, const int* in_sizes, int n_in,
                              void* d_out, int out_size, void* d_ws, size_t ws_size,
                              hipStream_t stream) {
    (void)in_sizes; (void)n_in; (void)out_size; (void)ws_size;

    const float* cls_prob  = (const float*)d_in[0];  // [B,R,NC]
    const float* rois      = (const float*)d_in[1];  // [B,R,5]
    const float* bbox_pred = (const float*)d_in[2];  // [B,R,4*NC]
    const float* im_info   = (const float*)d_in[3];  // [B,3]
    const float* thr       = (const float*)d_in[4];  // [NC]
    // d_in[5] = num_classes scalar (== NC, hardcoded)

    float* out  = (float*)d_out;                     // [B,NK,5] then [B] counts
    float* nout = out + (size_t)NB * NK * 5;

    // Workspace layout: sorted boxes then keep flags (~0.94 MB total).
    float* ws_box  = (float*)d_ws;                                   // B*NCLS*R*4 floats
    int*   ws_keep = (int*)((char*)d_ws +
                     (size_t)NB * NCLS * NR * 4 * sizeof(float));    // B*NCLS*R ints

    decode_rank_nms_kernel<<<NB * NCLS, 320, 0, stream>>>(
        cls_prob, rois, bbox_pred, im_info, thr, ws_box, ws_keep);

    select_topk_kernel<<<NB, 256, 0, stream>>>(ws_box, ws_keep, out, nout);
}